// SGN_57518202028479
// MI455X (gfx1250) — compile-verified
//
#include <hip/hip_runtime.h>

typedef float v2f __attribute__((ext_vector_type(2)));
typedef float v4f __attribute__((ext_vector_type(4)));
typedef float v8f __attribute__((ext_vector_type(8)));

#define LS 68                  // padded LDS row stride in floats (bank-conflict avoidance)
#define WAVES 2
#define THREADS (WAVES * 32)
#define TILE_M 32              // rows per wave (two 16-row WMMA tiles sharing B fragments)
#define ROWS_PER_BLOCK (WAVES * TILE_M)

struct MLPPtrs { const float* W[5]; const float* b[5]; };

__device__ __forceinline__ v8f wmma4(v2f a, v2f b, v8f c) {
  // D = A(16x4 f32) * B(4x16 f32) + C(16x16 f32)
  return __builtin_amdgcn_wmma_f32_16x16x4_f32(false, a, false, b, (short)0, c, false, false);
}

// Repacked weight layout (K padded to even with zeros):
//   Wq[(k>>1)*128 + (c&15)*8 + (c>>4)*2 + (k&1)] = W[k*64 + c]
// Lane n's four B fragments for k-pair p = two contiguous v4f loads at Wq + p*128 + n*8
// (16 lanes x 32B = 512B coalesced per half-wave).
__device__ __forceinline__ void load_bfrag(const float* __restrict__ Wq, int kpair, int n,
                                           v2f bf[4]) {
  const float* wrow = Wq + kpair * 128 + n * 8;
  const v4f w01 = *(const v4f*)(wrow);
  const v4f w23 = *(const v4f*)(wrow + 4);
  bf[0].x = w01.x; bf[0].y = w01.y;
  bf[1].x = w01.z; bf[1].y = w01.w;
  bf[2].x = w23.x; bf[2].y = w23.y;
  bf[3].x = w23.z; bf[3].y = w23.w;
}

// ---- per-wave 32x64 tile helpers ----

__device__ __forceinline__ void init_bias_m32(const float* __restrict__ bias,
                                              v8f acc0[4], v8f acc1[4], int lane) {
  const int n = lane & 15;
#pragma unroll
  for (int ct = 0; ct < 4; ++ct) {
    const float bv = bias[ct * 16 + n];
    v8f v;
#pragma unroll
    for (int i = 0; i < 8; ++i) v[i] = bv;
    acc0[ct] = v;
    acc1[ct] = v;
  }
}

// 64->64 matmul; A (32 rows) from per-wave LDS tile, Wq repacked global (WGP$/L2-resident)
__device__ __forceinline__ void compute64_m32(const float* __restrict__ Wq, const float* act,
                                              v8f acc0[4], v8f acc1[4], int lane) {
  const int r = lane & 15;
  const int khalf = (lane >> 4) << 1;
#pragma unroll 4
  for (int ks = 0; ks < 16; ++ks) {
    const int k0 = 4 * ks + khalf;
    const v2f a0 = *(const v2f*)(act + r * LS + k0);
    const v2f a1 = *(const v2f*)(act + (r + 16) * LS + k0);
    v2f bf[4];
    load_bfrag(Wq, k0 >> 1, r, bf);
#pragma unroll
    for (int ct = 0; ct < 4; ++ct) {
      acc0[ct] = wmma4(a0, bf[ct], acc0[ct]);
      acc1[ct] = wmma4(a1, bf[ct], acc1[ct]);
    }
  }
}

__device__ __forceinline__ void relu_store_lds_m32(v8f acc0[4], v8f acc1[4], float* out, int lane) {
  const int n = lane & 15;
  const int rbase = (lane >> 4) << 3;
#pragma unroll
  for (int ct = 0; ct < 4; ++ct)
#pragma unroll
    for (int i = 0; i < 8; ++i) {
      out[(rbase + i) * LS + ct * 16 + n] = fmaxf(acc0[ct][i], 0.0f);
      out[(16 + rbase + i) * LS + ct * 16 + n] = fmaxf(acc1[ct][i], 0.0f);
    }
}

__device__ __forceinline__ void store_global_m32(v8f acc0[4], v8f acc1[4],
                                                 float* __restrict__ out, int rowBase,
                                                 int nrows, int lane) {
  const int n = lane & 15;
  const int rbase = (lane >> 4) << 3;
#pragma unroll
  for (int ct = 0; ct < 4; ++ct)
#pragma unroll
    for (int i = 0; i < 8; ++i) {
      const int row0 = rowBase + rbase + i;
      const int row1 = row0 + 16;
      if (row0 < nrows) out[(size_t)row0 * 64 + ct * 16 + n] = acc0[ct][i];
      if (row1 < nrows) out[(size_t)row1 * 64 + ct * 16 + n] = acc1[ct][i];
    }
}

__device__ __forceinline__ void layer64_m32(const float* __restrict__ Wq,
                                            const float* __restrict__ bias,
                                            const float* in, float* out, int lane) {
  v8f a0[4], a1[4];
  init_bias_m32(bias, a0, a1, lane);
  compute64_m32(Wq, in, a0, a1, lane);
  relu_store_lds_m32(a0, a1, out, lane);
}

// dense first layer from global X [N, KIN]; rows clamped; K tail handled by zero-padded Wq rows
template <int KIN>
__device__ __forceinline__ void first_layer_dense_m32(const float* __restrict__ X,
                                                      const float* __restrict__ Wq,
                                                      v8f acc0[4], v8f acc1[4],
                                                      int rowBase, int N, int lane) {
  const int r = lane & 15;
  const int khalf = (lane >> 4) << 1;
  const int row0 = min(rowBase + r, N - 1);
  const int row1 = min(rowBase + 16 + r, N - 1);
  const float* x0 = X + (size_t)row0 * KIN;
  const float* x1 = X + (size_t)row1 * KIN;
  constexpr int KP = (KIN + 3) & ~3;
#pragma unroll
  for (int ks = 0; ks < KP / 4; ++ks) {
    const int k0 = 4 * ks + khalf;
    const int i0 = (k0 < KIN) ? k0 : (KIN - 1);      // clamped: in-bounds; tail zeroed via Wq pad
    const int i1 = (k0 + 1 < KIN) ? (k0 + 1) : (KIN - 1);
    v2f a0, a1;
    a0.x = x0[i0]; a0.y = x0[i1];
    a1.x = x1[i0]; a1.y = x1[i1];
    v2f bf[4];
    load_bfrag(Wq, k0 >> 1, r, bf);
#pragma unroll
    for (int ct = 0; ct < 4; ++ct) {
      acc0[ct] = wmma4(a0, bf[ct], acc0[ct]);
      acc1[ct] = wmma4(a1, bf[ct], acc1[ct]);
    }
  }
}

// ---- kernels ----

template <int KIN>
__global__ void enc_kernel(const float* __restrict__ X, MLPPtrs p, float* __restrict__ out, int N) {
  __shared__ float smem[WAVES * 2 * TILE_M * LS];
  const int wave = threadIdx.x >> 5, lane = threadIdx.x & 31;
  float* buf0 = smem + wave * (2 * TILE_M * LS);
  float* buf1 = buf0 + TILE_M * LS;
  const int rowBase = (blockIdx.x * WAVES + wave) * TILE_M;
  {
    v8f a0[4], a1[4];
    init_bias_m32(p.b[0], a0, a1, lane);
    first_layer_dense_m32<KIN>(X, p.W[0], a0, a1, rowBase, N, lane);
    relu_store_lds_m32(a0, a1, buf0, lane);
  }
  layer64_m32(p.W[1], p.b[1], buf0, buf1, lane);
  layer64_m32(p.W[2], p.b[2], buf1, buf0, lane);
  layer64_m32(p.W[3], p.b[3], buf0, buf1, lane);
  {
    v8f a0[4], a1[4];
    init_bias_m32(p.b[4], a0, a1, lane);
    compute64_m32(p.W[4], buf1, a0, a1, lane);
    store_global_m32(a0, a1, out, rowBase, N, lane);
  }
}

__global__ void msg_kernel(const float* __restrict__ h_v, const float* __restrict__ h_e,
                           const int* __restrict__ src, const int* __restrict__ dst,
                           MLPPtrs p, float* __restrict__ agg, int E) {
  __shared__ float smem[WAVES * 2 * TILE_M * LS];
  const int wave = threadIdx.x >> 5, lane = threadIdx.x & 31;
  float* buf0 = smem + wave * (2 * TILE_M * LS);
  float* buf1 = buf0 + TILE_M * LS;
  const int rowBase = (blockIdx.x * WAVES + wave) * TILE_M;
  const int r = lane & 15, n = r;
  const int khalf = (lane >> 4) << 1;
  // layer 1: gather-concat [h_v[src] | h_v[dst] | h_e], K = 192, 3 fixed-base segments
  {
    const int e0 = min(rowBase + r, E - 1);
    const int e1 = min(rowBase + 16 + r, E - 1);
    const float* seg0[3] = { h_v + (size_t)src[e0] * 64, h_v + (size_t)dst[e0] * 64,
                             h_e + (size_t)e0 * 64 };
    const float* seg1[3] = { h_v + (size_t)src[e1] * 64, h_v + (size_t)dst[e1] * 64,
                             h_e + (size_t)e1 * 64 };
    const float* W0 = p.W[0];
    v8f a0[4], a1[4];
    init_bias_m32(p.b[0], a0, a1, lane);
#pragma unroll
    for (int seg = 0; seg < 3; ++seg) {
      const float* q0 = seg0[seg];
      const float* q1 = seg1[seg];
#pragma unroll 4
      for (int ks = 0; ks < 16; ++ks) {
        const int kk = 4 * ks + khalf;
        const int kfull = seg * 64 + kk;
        const v2f av0 = *(const v2f*)(q0 + kk);
        const v2f av1 = *(const v2f*)(q1 + kk);
        v2f bf[4];
        load_bfrag(W0, kfull >> 1, n, bf);
#pragma unroll
        for (int ct = 0; ct < 4; ++ct) {
          a0[ct] = wmma4(av0, bf[ct], a0[ct]);
          a1[ct] = wmma4(av1, bf[ct], a1[ct]);
        }
      }
    }
    relu_store_lds_m32(a0, a1, buf0, lane);
  }
  layer64_m32(p.W[1], p.b[1], buf0, buf1, lane);
  layer64_m32(p.W[2], p.b[2], buf1, buf0, lane);
  layer64_m32(p.W[3], p.b[3], buf0, buf1, lane);
  // layer 5: linear, scatter-add into agg[dst]
  {
    v8f a0[4], a1[4];
    init_bias_m32(p.b[4], a0, a1, lane);
    compute64_m32(p.W[4], buf1, a0, a1, lane);
    const int rbase = (lane >> 4) << 3;
    int d0[8], d1[8];
#pragma unroll
    for (int i = 0; i < 8; ++i) {
      const int ea = rowBase + rbase + i;
      const int eb = ea + 16;
      d0[i] = (ea < E) ? dst[ea] : -1;
      d1[i] = (eb < E) ? dst[eb] : -1;
    }
#pragma unroll
    for (int ct = 0; ct < 4; ++ct)
#pragma unroll
      for (int i = 0; i < 8; ++i) {
        if (d0[i] >= 0) atomicAdd(&agg[(size_t)d0[i] * 64 + ct * 16 + n], a0[ct][i]);
        if (d1[i] >= 0) atomicAdd(&agg[(size_t)d1[i] * 64 + ct * 16 + n], a1[ct][i]);
      }
  }
}

__global__ void upd_kernel(const float* __restrict__ h_v, const float* __restrict__ agg,
                           const float* __restrict__ deg, MLPPtrs p,
                           float* __restrict__ h_out, int N) {
  __shared__ float smem[WAVES * 2 * TILE_M * LS];
  const int wave = threadIdx.x >> 5, lane = threadIdx.x & 31;
  float* buf0 = smem + wave * (2 * TILE_M * LS);
  float* buf1 = buf0 + TILE_M * LS;
  const int rowBase = (blockIdx.x * WAVES + wave) * TILE_M;
  const int r = lane & 15, n = r;
  const int khalf = (lane >> 4) << 1;
  // layer 1: concat [h_v | agg/deg], K = 128, two fixed-base segments
  {
    const int v0 = min(rowBase + r, N - 1);
    const int v1 = min(rowBase + 16 + r, N - 1);
    const float inv0 = 1.0f / fmaxf(deg[v0], 1.0f);
    const float inv1 = 1.0f / fmaxf(deg[v1], 1.0f);
    const float* q00 = h_v + (size_t)v0 * 64;
    const float* q01 = agg + (size_t)v0 * 64;
    const float* q10 = h_v + (size_t)v1 * 64;
    const float* q11 = agg + (size_t)v1 * 64;
    const float* W0 = p.W[0];
    v8f a0[4], a1[4];
    init_bias_m32(p.b[0], a0, a1, lane);
#pragma unroll 4
    for (int ks = 0; ks < 16; ++ks) {          // segment 0: h_v
      const int kk = 4 * ks + khalf;
      const v2f av0 = *(const v2f*)(q00 + kk);
      const v2f av1 = *(const v2f*)(q10 + kk);
      v2f bf[4];
      load_bfrag(W0, kk >> 1, n, bf);
#pragma unroll
      for (int ct = 0; ct < 4; ++ct) {
        a0[ct] = wmma4(av0, bf[ct], a0[ct]);
        a1[ct] = wmma4(av1, bf[ct], a1[ct]);
      }
    }
#pragma unroll 4
    for (int ks = 0; ks < 16; ++ks) {          // segment 1: agg * inv_deg
      const int kk = 4 * ks + khalf;
      v2f av0 = *(const v2f*)(q01 + kk);
      v2f av1 = *(const v2f*)(q11 + kk);
      av0.x *= inv0; av0.y *= inv0;
      av1.x *= inv1; av1.y *= inv1;
      v2f bf[4];
      load_bfrag(W0, (64 + kk) >> 1, n, bf);
#pragma unroll
      for (int ct = 0; ct < 4; ++ct) {
        a0[ct] = wmma4(av0, bf[ct], a0[ct]);
        a1[ct] = wmma4(av1, bf[ct], a1[ct]);
      }
    }
    relu_store_lds_m32(a0, a1, buf0, lane);
  }
  layer64_m32(p.W[1], p.b[1], buf0, buf1, lane);
  layer64_m32(p.W[2], p.b[2], buf1, buf0, lane);
  layer64_m32(p.W[3], p.b[3], buf0, buf1, lane);
  {
    v8f a0[4], a1[4];
    init_bias_m32(p.b[4], a0, a1, lane);
    compute64_m32(p.W[4], buf1, a0, a1, lane);
    store_global_m32(a0, a1, h_out, rowBase, N, lane);
  }
}

__global__ void dec_kernel(const float* __restrict__ h_v, MLPPtrs p,
                           float* __restrict__ out, int N) {
  __shared__ float smem[WAVES * 2 * TILE_M * LS];
  const int wave = threadIdx.x >> 5, lane = threadIdx.x & 31;
  float* buf0 = smem + wave * (2 * TILE_M * LS);
  float* buf1 = buf0 + TILE_M * LS;
  const int rowBase = (blockIdx.x * WAVES + wave) * TILE_M;
  const int r = lane & 15, n = r;
  const int khalf = (lane >> 4) << 1;
  {
    v8f a0[4], a1[4];
    init_bias_m32(p.b[0], a0, a1, lane);
    first_layer_dense_m32<64>(h_v, p.W[0], a0, a1, rowBase, N, lane);
    relu_store_lds_m32(a0, a1, buf0, lane);
  }
  layer64_m32(p.W[1], p.b[1], buf0, buf1, lane);
  layer64_m32(p.W[2], p.b[2], buf1, buf0, lane);
  layer64_m32(p.W[3], p.b[3], buf0, buf1, lane);
  // layer 5: 64 -> 3, original W [64,3] row-major; clamped loads + cndmask zeroing
  {
    const int rbase = (lane >> 4) << 3;
    const int nc = min(n, 2);
    const float bv = p.b[4][nc];
    v8f acc0, acc1;
#pragma unroll
    for (int i = 0; i < 8; ++i) { acc0[i] = bv; acc1[i] = bv; }
    const float* W4 = p.W[4];
#pragma unroll 4
    for (int ks = 0; ks < 16; ++ks) {
      const int k0 = 4 * ks + khalf;
      const v2f av0 = *(const v2f*)(buf1 + r * LS + k0);
      const v2f av1 = *(const v2f*)(buf1 + (r + 16) * LS + k0);
      const float wx = W4[k0 * 3 + nc];
      const float wy = W4[(k0 + 1) * 3 + nc];
      v2f b;
      b.x = (n < 3) ? wx : 0.0f;
      b.y = (n < 3) ? wy : 0.0f;
      acc0 = wmma4(av0, b, acc0);
      acc1 = wmma4(av1, b, acc1);
    }
#pragma unroll
    for (int i = 0; i < 8; ++i) {
      const int row0 = rowBase + rbase + i;
      const int row1 = row0 + 16;
      if (n < 3) {
        if (row0 < N) out[(size_t)row0 * 3 + n] = acc0[i];
        if (row1 < N) out[(size_t)row1 * 3 + n] = acc1[i];
      }
    }
  }
}

// ---- single-wave readout over 8 pooled rows (16-row tile, rows 8..15 zero) ----

__device__ __forceinline__ void layer64_16(const float* __restrict__ Wq,
                                           const float* __restrict__ bias,
                                           const float* in, float* out, int lane) {
  const int r = lane & 15;
  const int khalf = (lane >> 4) << 1;
  const int rbase = (lane >> 4) << 3;
  v8f acc[4];
#pragma unroll
  for (int ct = 0; ct < 4; ++ct) {
    const float bv = bias[ct * 16 + r];
#pragma unroll
    for (int i = 0; i < 8; ++i) acc[ct][i] = bv;
  }
#pragma unroll 4
  for (int ks = 0; ks < 16; ++ks) {
    const int k0 = 4 * ks + khalf;
    const v2f a = *(const v2f*)(in + r * LS + k0);
    v2f bf[4];
    load_bfrag(Wq, k0 >> 1, r, bf);
#pragma unroll
    for (int ct = 0; ct < 4; ++ct) acc[ct] = wmma4(a, bf[ct], acc[ct]);
  }
#pragma unroll
  for (int ct = 0; ct < 4; ++ct)
#pragma unroll
    for (int i = 0; i < 8; ++i)
      out[(rbase + i) * LS + ct * 16 + r] = fmaxf(acc[ct][i], 0.0f);
}

__global__ void readout_kernel(const float* __restrict__ psum, const float* __restrict__ pcnt,
                               MLPPtrs p, float* __restrict__ out) {
  __shared__ float smem[2 * 16 * LS];
  const int lane = threadIdx.x & 31;
  float* buf0 = smem;
  float* buf1 = smem + 16 * LS;
  const int r = lane & 15, n = r;
  const int khalf = (lane >> 4) << 1;
  // layer 1: A = pooled mean rows (graphs 0..7)
  {
    const bool rok = r < 8;
    const int rc = rok ? r : 0;
    const float inv = rok ? 1.0f / fmaxf(pcnt[rc], 1.0f) : 0.0f;
    const float* W0 = p.W[0];
    v8f acc[4];
#pragma unroll
    for (int ct = 0; ct < 4; ++ct) {
      const float bv = p.b[0][ct * 16 + n];
#pragma unroll
      for (int i = 0; i < 8; ++i) acc[ct][i] = bv;
    }
#pragma unroll 4
    for (int ks = 0; ks < 16; ++ks) {
      const int k0 = 4 * ks + khalf;
      v2f a = *(const v2f*)(psum + rc * 64 + k0);
      a.x *= inv; a.y *= inv;                   // inv==0 zeroes rows 8..15
      v2f bf[4];
      load_bfrag(W0, k0 >> 1, n, bf);
#pragma unroll
      for (int ct = 0; ct < 4; ++ct) acc[ct] = wmma4(a, bf[ct], acc[ct]);
    }
    const int rbase = (lane >> 4) << 3;
#pragma unroll
    for (int ct = 0; ct < 4; ++ct)
#pragma unroll
      for (int i = 0; i < 8; ++i)
        buf0[(rbase + i) * LS + ct * 16 + n] = fmaxf(acc[ct][i], 0.0f);
  }
  layer64_16(p.W[1], p.b[1], buf0, buf1, lane);
  layer64_16(p.W[2], p.b[2], buf1, buf0, lane);
  layer64_16(p.W[3], p.b[3], buf0, buf1, lane);
  // layer 5: 64 -> 1, original W [64,1]
  {
    const float* W4 = p.W[4];
    const float bv = p.b[4][0];
    v8f acc;
#pragma unroll
    for (int i = 0; i < 8; ++i) acc[i] = bv;
#pragma unroll 4
    for (int ks = 0; ks < 16; ++ks) {
      const int k0 = 4 * ks + khalf;
      const v2f a = *(const v2f*)(buf1 + r * LS + k0);
      const float wx = W4[k0];
      const float wy = W4[k0 + 1];
      v2f b;
      b.x = (n == 0) ? wx : 0.0f;
      b.y = (n == 0) ? wy : 0.0f;
      acc = wmma4(a, b, acc);
    }
    const int rbase = (lane >> 4) << 3;
#pragma unroll
    for (int i = 0; i < 8; ++i) {
      const int row = rbase + i;
      if (n == 0 && row < 8) out[row] = acc[i];
    }
  }
}

// ---- auxiliary kernels ----

__global__ void repack_w_kernel(const float* __restrict__ W, float* __restrict__ Wq,
                                int K, int Kpad) {
  const int i = blockIdx.x * blockDim.x + threadIdx.x;
  const int total = Kpad * 64;
  if (i >= total) return;
  const int k = i >> 6;
  const int c = i & 63;
  const float v = (k < K) ? W[k * 64 + c] : 0.0f;
  Wq[(k >> 1) * 128 + (c & 15) * 8 + (c >> 4) * 2 + (k & 1)] = v;
}

__global__ void zero_kernel(float* __restrict__ p, size_t nfloats) {
  size_t i = (size_t)blockIdx.x * blockDim.x + threadIdx.x;
  const size_t stride = (size_t)gridDim.x * blockDim.x;
  for (; i < nfloats; i += stride) p[i] = 0.0f;
}

__global__ void deg_kernel(const int* __restrict__ dst, float* __restrict__ deg, int E) {
  const int e = blockIdx.x * blockDim.x + threadIdx.x;
  if (e < E) atomicAdd(&deg[dst[e]], 1.0f);
}

__global__ void pool_kernel(const float* __restrict__ h_v, const int* __restrict__ batch,
                            float* __restrict__ psum, float* __restrict__ pcnt, int N) {
  __shared__ float ls[8 * 64 + 8];
  for (int i = threadIdx.x; i < 8 * 64 + 8; i += blockDim.x) ls[i] = 0.0f;
  __syncthreads();
  const size_t total = (size_t)N * 64;
  for (size_t i = (size_t)blockIdx.x * blockDim.x + threadIdx.x; i < total;
       i += (size_t)gridDim.x * blockDim.x) {
    const int v = (int)(i >> 6);
    const int k = (int)(i & 63);
    const int g = batch[v];
    atomicAdd(&ls[g * 64 + k], h_v[i]);
    if (k == 0) atomicAdd(&ls[512 + g], 1.0f);
  }
  __syncthreads();
  for (int i = threadIdx.x; i < 8 * 64; i += blockDim.x) atomicAdd(&psum[i], ls[i]);
  if (threadIdx.x < 8) atomicAdd(&pcnt[threadIdx.x], ls[512 + threadIdx.x]);
}

extern "C" void kernel_launch(void* const* d_in, const int* in_sizes, int n_in,
                              void* d_out, int out_size, void* d_ws, size_t ws_size,
                              hipStream_t stream) {
  (void)n_in; (void)out_size; (void)ws_size;
  const int N = 50000, E = 500000;

  MLPPtrs node_enc, edge_enc, dec, readout;
  MLPPtrs msg[4], upd[4];
  auto F = [&](MLPPtrs& m, int base) {
    for (int i = 0; i < 5; ++i) {
      m.W[i] = (const float*)d_in[base + 2 * i];
      m.b[i] = (const float*)d_in[base + 2 * i + 1];
    }
  };

  const float *x, *edge_attr;
  const int *eidx, *batch;
  const bool insertion_order = (in_sizes[0] == 350000);  // x first vs batch first (jax-sorted)
  if (insertion_order) {
    x = (const float*)d_in[0]; edge_attr = (const float*)d_in[1];
    eidx = (const int*)d_in[2]; batch = (const int*)d_in[3];
    F(node_enc, 4); F(edge_enc, 14);
    for (int l = 0; l < 4; ++l) { F(msg[l], 24 + 20 * l); F(upd[l], 34 + 20 * l); }
    F(dec, 104); F(readout, 114);
  } else {  // sorted keys: batch, edge_attr, edge_index, params{dec, edge_enc, mp, node_enc, readout}, x
    batch = (const int*)d_in[0]; edge_attr = (const float*)d_in[1];
    eidx = (const int*)d_in[2]; x = (const float*)d_in[123];
    F(dec, 3); F(edge_enc, 13);
    for (int l = 0; l < 4; ++l) { F(msg[l], 23 + 20 * l); F(upd[l], 33 + 20 * l); }
    F(node_enc, 103); F(readout, 113);
  }
  const int* src = eidx;
  const int* dstp = eidx + E;

  float* ws = (float*)d_ws;
  float* hvA = ws;                         // N*64
  float* hvB = hvA + (size_t)N * 64;       // N*64
  float* he  = hvB + (size_t)N * 64;       // E*64
  float* agg = he + (size_t)E * 64;        // N*64
  float* deg = agg + (size_t)N * 64;       // N
  float* psum = deg + N;                   // 512
  float* pcnt = psum + 512;                // 8
  float* wcur = pcnt + 8;                  // repacked weights (~280K floats)

  // repack a weight matrix [K,64] into fragment-packed layout (K padded to even w/ zeros)
  auto repack = [&](const float* W, int K) -> const float* {
    const int Kpad = (K + 1) & ~1;
    float* dstw = wcur;
    wcur += (size_t)Kpad * 64;
    const int total = Kpad * 64;
    repack_w_kernel<<<(total + 255) / 256, 256, 0, stream>>>(W, dstw, K, Kpad);
    return dstw;
  };
  auto repack_mlp = [&](MLPPtrs& m, int K0, bool last_is_64wide) {
    m.W[0] = repack(m.W[0], K0);
    m.W[1] = repack(m.W[1], 64);
    m.W[2] = repack(m.W[2], 64);
    m.W[3] = repack(m.W[3], 64);
    if (last_is_64wide) m.W[4] = repack(m.W[4], 64);  // else keep original narrow W4
  };
  repack_mlp(node_enc, 7, true);
  repack_mlp(edge_enc, 4, true);
  for (int l = 0; l < 4; ++l) { repack_mlp(msg[l], 192, true); repack_mlp(upd[l], 128, true); }
  repack_mlp(dec, 64, false);
  repack_mlp(readout, 64, false);

  const int nb_n = (N + ROWS_PER_BLOCK - 1) / ROWS_PER_BLOCK;
  const int nb_e = (E + ROWS_PER_BLOCK - 1) / ROWS_PER_BLOCK;

  // degrees (constant across MP layers)
  zero_kernel<<<256, 256, 0, stream>>>(deg, (size_t)N);
  deg_kernel<<<(E + 255) / 256, 256, 0, stream>>>(dstp, deg, E);

  // encoders
  enc_kernel<7><<<nb_n, THREADS, 0, stream>>>(x, node_enc, hvA, N);
  enc_kernel<4><<<nb_e, THREADS, 0, stream>>>(edge_attr, edge_enc, he, E);

  // message passing
  float* cur = hvA;
  float* nxt = hvB;
  for (int l = 0; l < 4; ++l) {
    zero_kernel<<<512, 256, 0, stream>>>(agg, (size_t)N * 64);
    msg_kernel<<<nb_e, THREADS, 0, stream>>>(cur, he, src, dstp, msg[l], agg, E);
    upd_kernel<<<nb_n, THREADS, 0, stream>>>(cur, agg, deg, upd[l], nxt, N);
    float* t = cur; cur = nxt; nxt = t;
  }

  // node decoder -> d_out[0 .. N*3)
  float* out = (float*)d_out;
  dec_kernel<<<nb_n, THREADS, 0, stream>>>(cur, dec, out, N);

  // global readout -> d_out[N*3 .. N*3+8)
  zero_kernel<<<4, 256, 0, stream>>>(psum, (size_t)(512 + 8));
  pool_kernel<<<256, 256, 0, stream>>>(cur, batch, psum, pcnt, N);
  readout_kernel<<<1, 32, 0, stream>>>(psum, pcnt, readout, out + (size_t)N * 3);
}